// LMHMamba_Block1_17970143167022
// MI455X (gfx1250) — compile-verified
//
#include <hip/hip_runtime.h>
#include <math.h>

// ---------------- problem constants ----------------
#define B_   16
#define D_   256
#define H_   96
#define W_   96
#define HW_  (H_*W_)
#define DS_  64
#define MH_  256
#define S_   64
#define H2_  48
#define W2_  48
#define HW2_ (H2_*W2_)
#define H3_  32
#define HW3_ (H3_*H3_)

typedef __attribute__((ext_vector_type(2))) float v2f;
typedef __attribute__((ext_vector_type(8))) float v8f;

__device__ __forceinline__ float gelu_f(float v){ return 0.5f*v*(1.0f+erff(v*0.70710678118f)); }
__device__ __forceinline__ float sigm_f(float v){ return 1.0f/(1.0f+__expf(-v)); }

// =====================================================================
// Generic WMMA f32 pointwise-conv / batched GEMM:
//   out[b, o, t] = epilogue( sum_k W[b, o, k] * in[b, k, t] )
// M = out-channels (16/wave), N = 16 contiguous tokens (coalesced),
// K = in-channels, stepped 4 at a time through v_wmma_f32_16x16x4_f32.
//   act: 0=none 1=gelu 2=relu6   resmode: 0=none 1=add 2=mul
// =====================================================================
__global__ __launch_bounds__(128) void k_pw_gemm(
    const float* __restrict__ in, const float* __restrict__ Wt,
    const float* __restrict__ bias, const float* __restrict__ scale,
    const float* __restrict__ shift, const float* __restrict__ resid,
    float* __restrict__ out,
    int Cin, int Cout, int HWloc,
    long in_bstride, long out_bstride, long resid_bstride, long w_bstride,
    int act, int resmode)
{
  const int tilesPerB = HWloc >> 4;
  const int b    = blockIdx.x / tilesPerB;
  const int hw0  = (blockIdx.x % tilesPerB) << 4;
  const int wave = threadIdx.x >> 5;
  const int lane = threadIdx.x & 31;
  const int obase = blockIdx.y * 64 + wave * 16;
  if (obase >= Cout) return;                 // wave-uniform: EXEC stays all-1s
  const int lhi = lane >> 4;                 // half-wave selector
  const int lin = lane & 15;

  const float* __restrict__ inb  = in + (long)b * in_bstride + hw0;
  const float* __restrict__ Arow = Wt + (long)b * w_bstride + (long)(obase + lin) * Cin;

  v8f acc = {0.f,0.f,0.f,0.f,0.f,0.f,0.f,0.f};
  for (int k0 = 0; k0 < Cin; k0 += 4) {
    const int ka = k0 + (lhi << 1);          // lanes 0-15: K 0,1 | lanes 16-31: K 2,3
    v2f a; a.x = Arow[ka]; a.y = Arow[ka + 1];
    const float* __restrict__ Bp = inb + (long)ka * HWloc + lin;
    // locality=3 -> WGP-scope prefetch: pulls into ALL cache levels on miss
    // (SYS/DEV scope would bypass the WGP cache per the ISA prefetch table)
    __builtin_prefetch(Bp + 4 * HWloc, 0, 3);
    v2f bf; bf.x = Bp[0]; bf.y = Bp[HWloc];
    acc = __builtin_amdgcn_wmma_f32_16x16x4_f32(false, a, false, bf,
                                                (short)0, acc, false, false);
  }

  // C layout: VGPR r, lanes0-15 -> M=r, lanes16-31 -> M=r+8; N = lane&15
  #pragma unroll
  for (int r = 0; r < 8; r++) {
    const int o = obase + r + (lhi << 3);
    float v = acc[r];
    if (bias)  v += bias[o];
    if (scale) v = v * scale[o] + shift[o];
    if (act == 1)      v = gelu_f(v);
    else if (act == 2) v = fminf(fmaxf(v, 0.f), 6.f);
    const long p = (long)o * HWloc + hw0 + lin;
    if (resmode == 1)      v += resid[(long)b * resid_bstride + p];
    else if (resmode == 2) v *= resid[(long)b * resid_bstride + p];
    out[(long)b * out_bstride + p] = v;
  }
}

// =====================================================================
// Generic depthwise KxK conv (runtime K/dilation/pad) + bias + optional BN
// =====================================================================
__global__ void k_dw(const float* __restrict__ in, const float* __restrict__ w,
                     const float* __restrict__ bias, const float* __restrict__ scale,
                     const float* __restrict__ shift, float* __restrict__ out,
                     int C, int Hh, int Ww, long in_bstride, long out_bstride,
                     int K, int dil, int pad)
{
  long idx = (long)blockIdx.x * blockDim.x + threadIdx.x;
  long tot = (long)B_ * C * Hh * Ww;
  if (idx >= tot) return;
  int xx = idx % Ww; long t = idx / Ww;
  int yy = t % Hh;  t /= Hh;
  int c  = t % C;   int b = (int)(t / C);
  const float* ip = in + (long)b * in_bstride + (long)c * Hh * Ww;
  const float* wp = w + (long)c * K * K;
  float acc = 0.f;
  for (int ky = 0; ky < K; ky++) {
    int y2 = yy + ky * dil - pad;
    if ((unsigned)y2 >= (unsigned)Hh) continue;
    for (int kx = 0; kx < K; kx++) {
      int x2 = xx + kx * dil - pad;
      if ((unsigned)x2 >= (unsigned)Ww) continue;
      acc += wp[ky * K + kx] * ip[y2 * Ww + x2];
    }
  }
  if (bias)  acc += bias[c];
  if (scale) acc = acc * scale[c] + shift[c];
  out[(long)b * out_bstride + (long)c * Hh * Ww + yy * Ww + xx] = acc;
}

// =====================================================================
// Branch 2: fused maxpool3(s1,p1) + blurpool(filt4, stride3, reflect pad)
// =====================================================================
__device__ __forceinline__ int refl96(int r){ return r < 0 ? -r : (r >= H_ ? 2*(H_-1) - r : r); }
__device__ __forceinline__ float mp3v(const float* __restrict__ p, int y, int x){
  float m = -1e30f;
  for (int dy = -1; dy <= 1; dy++) { int y2 = y + dy; if ((unsigned)y2 >= (unsigned)H_) continue;
    for (int dx = -1; dx <= 1; dx++) { int x2 = x + dx; if ((unsigned)x2 >= (unsigned)W_) continue;
      m = fmaxf(m, p[y2 * W_ + x2]); } }
  return m;
}
__global__ void k_mra_pool(const float* __restrict__ x, float* __restrict__ xt)
{
  long idx = (long)blockIdx.x * blockDim.x + threadIdx.x;
  if (idx >= (long)B_ * DS_ * HW3_) return;
  int ox = idx % H3_; long t = idx / H3_;
  int oy = t % H3_;  t /= H3_;
  int c  = t % DS_;  int b = (int)(t / DS_);
  const float* p = x + ((long)b * D_ + DS_ + c) * HW_;
  const float fw[4] = {1.f, 3.f, 3.f, 1.f};
  float acc = 0.f;
  for (int fy = 0; fy < 4; fy++) { int ry = refl96(3*oy + fy - 1);
    for (int fx = 0; fx < 4; fx++) { int rx = refl96(3*ox + fx - 1);
      acc += fw[fy] * fw[fx] * mp3v(p, ry, rx); } }
  xt[((long)b * DS_ + c) * HW3_ + oy * H3_ + ox] = acc * (1.f / 64.f);
}

// =====================================================================
// Branch 2: four strip convs (incl. shear-collapsed diagonals) + BN + sigmoid
// =====================================================================
__global__ void k_mra_att(const float* __restrict__ xt,
                          const float* __restrict__ h1w, const float* __restrict__ v1w,
                          const float* __restrict__ h2w, const float* __restrict__ v2w,
                          const float* __restrict__ ms,  const float* __restrict__ mb,
                          float* __restrict__ gatt)
{
  long idx = (long)blockIdx.x * blockDim.x + threadIdx.x;
  if (idx >= (long)B_ * DS_ * HW3_) return;
  int j = idx % H3_; long t = idx / H3_;
  int i = t % H3_;  t /= H3_;
  int c = t % DS_;  int b = (int)(t / DS_);
  const float* p = xt + ((long)b * DS_ + c) * HW3_;
  float acc = 0.f;
  for (int a = 0; a < 11; a++) { int yy = i + a - 5; if ((unsigned)yy >= 32u) continue;
    for (int d = 0; d < 3; d++) {                           // xh1: 11x3, pad(5,1)
      int xx = j + d - 1; if ((unsigned)xx >= 32u) continue;
      acc += h1w[c*33 + a*3 + d] * p[yy*32 + xx]; } }
  for (int a = 0; a < 3; a++) { int yy = i + a - 1; if ((unsigned)yy >= 32u) continue;
    for (int d = 0; d < 11; d++) {                          // xw1: 3x11, pad(1,5)
      int xx = j + d - 5; if ((unsigned)xx >= 32u) continue;
      acc += v1w[c*33 + a*11 + d] * p[yy*32 + xx]; } }
  for (int a = 0; a < 11; a++) { int yy = i + a - 5; if ((unsigned)yy >= 32u) continue;
    for (int d = 0; d < 3; d++) {                           // xh2: diagonal shear collapsed
      int xx = j - (a - 5) + (d - 1); if ((unsigned)xx >= 32u) continue;
      acc += h2w[c*33 + a*3 + d] * p[yy*32 + xx]; } }
  for (int pp = 0; pp < 3; pp++) for (int q = 0; q < 11; q++) {   // xw2: anti-diagonal
    int yy = i + (pp - 1) - (q - 5); int xx = j + (q - 5);
    if ((unsigned)yy >= 32u || (unsigned)xx >= 32u) continue;
    acc += v2w[c*33 + pp*11 + q] * p[yy*32 + xx]; }
  gatt[((long)b * DS_ + c) * HW3_ + i * 32 + j] = sigm_f(ms[c] * acc + mb[c]);
}

__global__ void k_mra_apply(const float* __restrict__ x, const float* __restrict__ gatt,
                            float* __restrict__ xa)
{
  long idx = (long)blockIdx.x * blockDim.x + threadIdx.x;
  if (idx >= (long)B_ * DS_ * HW_) return;
  int w = idx % W_; long t = idx / W_;
  int h = t % H_;  t /= H_;
  int c = t % DS_; int b = (int)(t / DS_);
  float g = gatt[((long)b * DS_ + c) * HW3_ + (h / 3) * 32 + (w / 3)];
  long src = ((long)b * D_ + DS_ + c) * HW_ + h * W_ + w;
  xa[src] = x[src] * g;
}

// =====================================================================
// Branch 3: row softmax stats over L (A_p is constant per row -> cancels)
// =====================================================================
__global__ __launch_bounds__(256) void k_smax(const float* __restrict__ bcdt2,
                                              float* __restrict__ stats)
{
  int row = blockIdx.x;                  // b*S + s
  int b = row / S_, s = row % S_;
  const float* dt = bcdt2 + ((long)b * (3 * S_) + 2 * S_ + s) * HW_;
  __shared__ float red[256];
  float m = -1e30f;
  for (int i = threadIdx.x; i < HW_; i += 256) m = fmaxf(m, dt[i]);
  red[threadIdx.x] = m; __syncthreads();
  for (int st = 128; st > 0; st >>= 1) {
    if (threadIdx.x < st) red[threadIdx.x] = fmaxf(red[threadIdx.x], red[threadIdx.x + st]);
    __syncthreads();
  }
  m = red[0]; __syncthreads();
  float sum = 0.f;
  for (int i = threadIdx.x; i < HW_; i += 256) sum += __expf(dt[i] - m);
  red[threadIdx.x] = sum; __syncthreads();
  for (int st = 128; st > 0; st >>= 1) {
    if (threadIdx.x < st) red[threadIdx.x] += red[threadIdx.x + st];
    __syncthreads();
  }
  if (threadIdx.x == 0) { stats[2 * row] = m; stats[2 * row + 1] = 1.f / red[0]; }
}

// Build AB = softmax(dt)*Bm, transposed to (B, L, S) via LDS tile transpose
__global__ __launch_bounds__(256) void k_ab(const float* __restrict__ bcdt2,
                                            const float* __restrict__ stats,
                                            float* __restrict__ abuf)
{
  int b = blockIdx.z;
  int l0 = blockIdx.x * 16, s0 = blockIdx.y * 16;
  int tx = threadIdx.x & 15, ty = threadIdx.x >> 4;
  int s = s0 + ty, l = l0 + tx;
  float mx  = stats[2 * (b * S_ + s)];
  float ism = stats[2 * (b * S_ + s) + 1];
  float dt  = bcdt2[((long)b * 192 + 128 + s) * HW_ + l];
  float Bm  = bcdt2[((long)b * 192 + s) * HW_ + l];
  __shared__ float tile[16][17];
  tile[ty][tx] = __expf(dt - mx) * ism * Bm;
  __syncthreads();
  abuf[((long)b * HW_ + (l0 + ty)) * S_ + (s0 + tx)] = tile[tx][ty];
}

__global__ void k_hz_gate(const float* __restrict__ hz, const float* __restrict__ D_p,
                          float* __restrict__ h2)
{
  long idx = (long)blockIdx.x * blockDim.x + threadIdx.x;
  if (idx >= (long)B_ * DS_ * S_) return;
  int s = idx % S_; long t = idx / S_;
  int c = t % DS_; int b = (int)(t / DS_);
  float h1 = hz[((long)b * 2 * DS_ + c) * S_ + s];
  float z  = hz[((long)b * 2 * DS_ + DS_ + c) * S_ + s];
  h2[idx] = h1 * (z * sigm_f(z)) + h1 * D_p[0];
}

// =====================================================================
// Branch 4 helpers
// =====================================================================
__global__ void k_pool2(const float* __restrict__ x, float* __restrict__ v, int* __restrict__ idx)
{
  long t0 = (long)blockIdx.x * blockDim.x + threadIdx.x;
  if (t0 >= (long)B_ * DS_ * HW2_) return;
  int j = t0 % W2_; long t = t0 / W2_;
  int i = t % H2_; t /= H2_;
  int c = t % DS_; int b = (int)(t / DS_);
  const float* p = x + ((long)b * D_ + 3 * DS_ + c) * HW_;
  float best = -1e30f; int bi = 0;
  for (int d = 0; d < 4; d++) {
    float u = p[(2*i + (d >> 1)) * W_ + 2*j + (d & 1)];
    if (u > best) { best = u; bi = d; }
  }
  v[t0] = best; idx[t0] = bi;
}

__global__ void k_avgmax(const float* __restrict__ a1n, const float* __restrict__ a2n,
                         float* __restrict__ am)
{
  long idx = (long)blockIdx.x * blockDim.x + threadIdx.x;
  if (idx >= (long)B_ * HW2_) return;
  int pix = idx % HW2_; int b = (int)(idx / HW2_);
  float s = 0.f, m = -1e30f;
  for (int c = 0; c < 32; c++) { float u = a1n[((long)b*32 + c)*HW2_ + pix]; s += u; m = fmaxf(m, u); }
  for (int c = 0; c < 32; c++) { float u = a2n[((long)b*32 + c)*HW2_ + pix]; s += u; m = fmaxf(m, u); }
  am[((long)b * 2) * HW2_ + pix]     = s * (1.f / 64.f);
  am[((long)b * 2 + 1) * HW2_ + pix] = m;
}

__global__ void k_sq(const float* __restrict__ am, const float* __restrict__ sqw,
                     const float* __restrict__ sqb, float* __restrict__ sig)
{
  long idx = (long)blockIdx.x * blockDim.x + threadIdx.x;
  if (idx >= (long)B_ * 2 * HW2_) return;
  int xx = idx % W2_; long t = idx / W2_;
  int yy = t % H2_; t /= H2_;
  int o  = t % 2;   int b = (int)(t / 2);
  float acc = sqb[o];
  for (int ic = 0; ic < 2; ic++)
    for (int ky = 0; ky < 7; ky++) { int y2 = yy + ky - 3; if ((unsigned)y2 >= 48u) continue;
      for (int kx = 0; kx < 7; kx++) { int x2 = xx + kx - 3; if ((unsigned)x2 >= 48u) continue;
        acc += sqw[((o*2 + ic)*7 + ky)*7 + kx] * am[((long)b*2 + ic)*HW2_ + y2*W2_ + x2]; } }
  sig[((long)b * 2 + o) * HW2_ + yy * W2_ + xx] = sigm_f(acc);
}

__global__ void k_mix(const float* __restrict__ a1n, const float* __restrict__ a2n,
                      const float* __restrict__ sig, float* __restrict__ attnv)
{
  long idx = (long)blockIdx.x * blockDim.x + threadIdx.x;
  if (idx >= (long)B_ * 32 * HW2_) return;
  int pix = idx % HW2_; long t = idx / HW2_;
  int b = (int)(t / 32);
  float s0 = sig[((long)b * 2) * HW2_ + pix];
  float s1 = sig[((long)b * 2 + 1) * HW2_ + pix];
  attnv[idx] = a1n[idx] * s0 + a2n[idx] * s1;
}

__global__ void k_unpool_bn(const float* __restrict__ x, const float* __restrict__ p2o,
                            const int* __restrict__ idx, const float* __restrict__ gs,
                            const float* __restrict__ gb, float* __restrict__ xa)
{
  long t0 = (long)blockIdx.x * blockDim.x + threadIdx.x;
  if (t0 >= (long)B_ * DS_ * HW_) return;
  int w = t0 % W_; long t = t0 / W_;
  int h = t % H_; t /= H_;
  int c = t % DS_; int b = (int)(t / DS_);
  int i = h >> 1, j = w >> 1, sub = ((h & 1) << 1) | (w & 1);
  long lo = ((long)b * DS_ + c) * HW2_ + i * W2_ + j;
  float u = (idx[lo] == sub) ? p2o[lo] : 0.f;
  long hi = ((long)b * D_ + 3 * DS_ + c) * HW_ + h * W_ + w;
  xa[hi] = gs[c] * (x[hi] + u) + gb[c];
}

// =====================================================================
extern "C" void kernel_launch(void* const* d_in, const int* in_sizes, int n_in,
                              void* d_out, int out_size, void* d_ws, size_t ws_size,
                              hipStream_t stream)
{
  (void)in_sizes; (void)n_in; (void)out_size; (void)ws_size;
  const float* x      = (const float*)d_in[0];
  const float* dw1_w  = (const float*)d_in[1];
  const float* dw1_b  = (const float*)d_in[2];
  const float* dw1_s  = (const float*)d_in[3];
  const float* dw1_bb = (const float*)d_in[4];
  const float* f1_w   = (const float*)d_in[5];
  const float* f1_b   = (const float*)d_in[6];
  const float* f2_w   = (const float*)d_in[7];
  const float* f2_b   = (const float*)d_in[8];
  const float* g_w    = (const float*)d_in[9];
  const float* g_b    = (const float*)d_in[10];
  const float* g_s    = (const float*)d_in[11];
  const float* g_bb   = (const float*)d_in[12];
  const float* dw2_w  = (const float*)d_in[13];
  const float* dw2_b  = (const float*)d_in[14];
  const float* h1_w   = (const float*)d_in[15];
  const float* v1_w   = (const float*)d_in[16];
  const float* h2_w   = (const float*)d_in[17];
  const float* v2_w   = (const float*)d_in[18];
  const float* mra_s  = (const float*)d_in[19];
  const float* mra_b  = (const float*)d_in[20];
  const float* bcdt_w = (const float*)d_in[21];
  const float* dwm_w  = (const float*)d_in[22];
  const float* dwm_b  = (const float*)d_in[23];
  const float* hz_w   = (const float*)d_in[24];
  const float* out_w  = (const float*)d_in[25];
  const float* D_p    = (const float*)d_in[27];   // A_p (26) cancels in softmax
  const float* p1_w   = (const float*)d_in[28];
  const float* p1_b   = (const float*)d_in[29];
  const float* c0_w   = (const float*)d_in[30];
  const float* c0_b   = (const float*)d_in[31];
  const float* cs_w   = (const float*)d_in[32];
  const float* cs_b   = (const float*)d_in[33];
  const float* c1_w   = (const float*)d_in[34];
  const float* c1_b   = (const float*)d_in[35];
  const float* c2_w   = (const float*)d_in[36];
  const float* c2_b   = (const float*)d_in[37];
  const float* sq_w   = (const float*)d_in[38];
  const float* sq_b   = (const float*)d_in[39];
  const float* cv_w   = (const float*)d_in[40];
  const float* cv_b   = (const float*)d_in[41];
  const float* p2_w   = (const float*)d_in[42];
  const float* p2_b   = (const float*)d_in[43];
  const float* ga_s   = (const float*)d_in[44];
  const float* ga_b   = (const float*)d_in[45];
  const float* mlp1_w = (const float*)d_in[46];
  const float* mlp_s  = (const float*)d_in[47];
  const float* mlp_b  = (const float*)d_in[48];
  const float* mlp2_w = (const float*)d_in[49];
  const float* n1_s   = (const float*)d_in[50];
  const float* n1_b   = (const float*)d_in[51];
  float* outp = (float*)d_out;

  // ---- workspace carve (floats) ----
  float* ws = (float*)d_ws;
  size_t off = 0;
  auto alloc = [&](size_t n) { float* p = ws + off; off += n; return p; };
  float* x1o    = alloc((size_t)B_ * DS_ * HW_);
  float* x1mid  = alloc((size_t)B_ * DS_ * HW_);
  float* xa     = alloc((size_t)B_ * D_ * HW_);
  float* m1     = alloc((size_t)B_ * MH_ * HW_);   // also reused as branch-1 gate temp
  float* xt     = alloc((size_t)B_ * DS_ * HW3_);
  float* gatt   = alloc((size_t)B_ * DS_ * HW3_);
  float* bcdtA  = alloc((size_t)B_ * 3 * S_ * HW_);
  float* bcdtB  = alloc((size_t)B_ * 3 * S_ * HW_);
  float* stats  = alloc((size_t)2 * B_ * S_);
  float* abuf   = alloc((size_t)B_ * HW_ * S_);
  float* hbuf   = alloc((size_t)B_ * DS_ * S_);
  float* hzbuf  = alloc((size_t)B_ * 2 * DS_ * S_);
  float* hobuf  = alloc((size_t)B_ * DS_ * S_);
  float* v4     = alloc((size_t)B_ * DS_ * HW2_);
  int*   idx4   = (int*)alloc((size_t)B_ * DS_ * HW2_);
  float* xg4    = alloc((size_t)B_ * DS_ * HW2_);
  float* a1buf  = alloc((size_t)B_ * DS_ * HW2_);
  float* a2buf  = alloc((size_t)B_ * DS_ * HW2_);
  float* a1n    = alloc((size_t)B_ * 32 * HW2_);
  float* a2n    = alloc((size_t)B_ * 32 * HW2_);
  float* amean  = alloc((size_t)B_ * 2 * HW2_);
  float* sig    = alloc((size_t)B_ * 2 * HW2_);
  float* attnv  = alloc((size_t)B_ * 32 * HW2_);

  auto eg = [](long n) { return dim3((unsigned)((n + 255) / 256)); };
  auto pw = [&](const float* in, const float* Wt, const float* bias,
                const float* scale, const float* shift, const float* resid,
                float* out, int Cin, int Cout, int HWloc,
                long inb, long outb, long resb, long wb, int act, int resmode) {
    dim3 g((unsigned)(B_ * (HWloc / 16)), (unsigned)((Cout + 63) / 64));
    k_pw_gemm<<<g, dim3(128), 0, stream>>>(in, Wt, bias, scale, shift, resid, out,
                                           Cin, Cout, HWloc, inb, outb, resb, wb,
                                           act, resmode);
  };

  // ---------------- branch 1 ----------------
  k_dw<<<eg((long)B_*DS_*HW_), 256, 0, stream>>>(x, dw1_w, dw1_b, dw1_s, dw1_bb, x1o,
      DS_, H_, W_, (long)D_*HW_, (long)DS_*HW_, 7, 1, 3);
  pw(x1o, f1_w, f1_b, nullptr, nullptr, nullptr, m1, DS_, MH_, HW_,
     (long)DS_*HW_, (long)MH_*HW_, 0, 0, /*relu6*/2, 0);
  pw(x1o, f2_w, f2_b, nullptr, nullptr, m1, m1, DS_, MH_, HW_,
     (long)DS_*HW_, (long)MH_*HW_, (long)MH_*HW_, 0, 0, /*mul gate*/2);
  pw(m1, g_w, g_b, g_s, g_bb, nullptr, x1mid, MH_, DS_, HW_,
     (long)MH_*HW_, (long)DS_*HW_, 0, 0, 0, 0);
  k_dw<<<eg((long)B_*DS_*HW_), 256, 0, stream>>>(x1mid, dw2_w, dw2_b, nullptr, nullptr,
      xa /*ch 0..63*/, DS_, H_, W_, (long)DS_*HW_, (long)D_*HW_, 7, 1, 3);

  // ---------------- branch 2 (MRA) ----------------
  k_mra_pool<<<eg((long)B_*DS_*HW3_), 256, 0, stream>>>(x, xt);
  k_mra_att<<<eg((long)B_*DS_*HW3_), 256, 0, stream>>>(xt, h1_w, v1_w, h2_w, v2_w,
                                                       mra_s, mra_b, gatt);
  k_mra_apply<<<eg((long)B_*DS_*HW_), 256, 0, stream>>>(x, gatt, xa + (long)DS_*HW_);

  // ---------------- branch 3 (HSM-SSD) ----------------
  pw(x + (long)2*DS_*HW_, bcdt_w, nullptr, nullptr, nullptr, nullptr, bcdtA,
     DS_, 3*S_, HW_, (long)D_*HW_, (long)3*S_*HW_, 0, 0, 0, 0);
  k_dw<<<eg((long)B_*3*S_*HW_), 256, 0, stream>>>(bcdtA, dwm_w, dwm_b, nullptr, nullptr,
      bcdtB, 3*S_, H_, W_, (long)3*S_*HW_, (long)3*S_*HW_, 3, 1, 1);
  k_smax<<<dim3(B_*S_), 256, 0, stream>>>(bcdtB, stats);
  k_ab<<<dim3(HW_/16, S_/16, B_), 256, 0, stream>>>(bcdtB, stats, abuf);
  // h = x3 · ABᵀ : per-batch weights = x3 rows (stride HW), K = L = 9216
  pw(abuf, x + (long)2*DS_*HW_, nullptr, nullptr, nullptr, nullptr, hbuf,
     HW_, DS_, S_, (long)HW_*S_, (long)DS_*S_, 0, (long)D_*HW_, 0, 0);
  pw(hbuf, hz_w, nullptr, nullptr, nullptr, nullptr, hzbuf, DS_, 2*DS_, S_,
     (long)DS_*S_, (long)2*DS_*S_, 0, 0, 0, 0);
  k_hz_gate<<<eg((long)B_*DS_*S_), 256, 0, stream>>>(hzbuf, D_p, hbuf);
  pw(hbuf, out_w, nullptr, nullptr, nullptr, nullptr, hobuf, DS_, DS_, S_,
     (long)DS_*S_, (long)DS_*S_, 0, 0, 0, 0);
  // y = ho · C : per-batch weights hobuf, activations Cm = bcdtB channels 64..127
  pw(bcdtB + (long)S_*HW_, hobuf, nullptr, nullptr, nullptr, nullptr,
     xa + (long)2*DS_*HW_, S_, DS_, HW_, (long)3*S_*HW_, (long)D_*HW_, 0,
     (long)DS_*S_, 0, 0);

  // ---------------- branch 4 (GA12) ----------------
  k_pool2<<<eg((long)B_*DS_*HW2_), 256, 0, stream>>>(x, v4, idx4);
  pw(v4, p1_w, p1_b, nullptr, nullptr, nullptr, xg4, DS_, DS_, HW2_,
     (long)DS_*HW2_, (long)DS_*HW2_, 0, 0, /*gelu*/1, 0);
  k_dw<<<eg((long)B_*DS_*HW2_), 256, 0, stream>>>(xg4, c0_w, c0_b, nullptr, nullptr,
      a1buf, DS_, H2_, W2_, (long)DS_*HW2_, (long)DS_*HW2_, 5, 1, 2);
  k_dw<<<eg((long)B_*DS_*HW2_), 256, 0, stream>>>(a1buf, cs_w, cs_b, nullptr, nullptr,
      a2buf, DS_, H2_, W2_, (long)DS_*HW2_, (long)DS_*HW2_, 7, 3, 9);
  pw(a1buf, c1_w, c1_b, nullptr, nullptr, nullptr, a1n, DS_, 32, HW2_,
     (long)DS_*HW2_, (long)32*HW2_, 0, 0, 0, 0);
  pw(a2buf, c2_w, c2_b, nullptr, nullptr, nullptr, a2n, DS_, 32, HW2_,
     (long)DS_*HW2_, (long)32*HW2_, 0, 0, 0, 0);
  k_avgmax<<<eg((long)B_*HW2_), 256, 0, stream>>>(a1n, a2n, amean);
  k_sq<<<eg((long)B_*2*HW2_), 256, 0, stream>>>(amean, sq_w, sq_b, sig);
  k_mix<<<eg((long)B_*32*HW2_), 256, 0, stream>>>(a1n, a2n, sig, attnv);
  // cv(attn) then fused x_*attn via multiplicative residual
  pw(attnv, cv_w, cv_b, nullptr, nullptr, xg4, a1buf, 32, DS_, HW2_,
     (long)32*HW2_, (long)DS_*HW2_, (long)DS_*HW2_, 0, 0, /*mul*/2);
  pw(a1buf, p2_w, p2_b, nullptr, nullptr, nullptr, a2buf, DS_, DS_, HW2_,
     (long)DS_*HW2_, (long)DS_*HW2_, 0, 0, 0, 0);
  k_unpool_bn<<<eg((long)B_*DS_*HW_), 256, 0, stream>>>(x, a2buf, idx4, ga_s, ga_b, xa);

  // ---------------- final MLP + residual ----------------
  pw(xa, mlp1_w, nullptr, mlp_s, mlp_b, nullptr, m1, D_, MH_, HW_,
     (long)D_*HW_, (long)MH_*HW_, 0, 0, /*gelu*/1, 0);
  pw(m1, mlp2_w, nullptr, n1_s, n1_b, x, outp, MH_, D_, HW_,
     (long)MH_*HW_, (long)D_*HW_, (long)D_*HW_, 0, 0, /*add*/1);
}